// NettackSurrogate_83580063580546
// MI455X (gfx1250) — compile-verified
//
#include <hip/hip_runtime.h>

typedef __attribute__((ext_vector_type(2))) float v2f;
typedef __attribute__((ext_vector_type(8))) float v8f;

// ---------------- utility fills ----------------
__global__ void fill_kernel(float* __restrict__ p, float v, long long n) {
    long long i = (long long)blockIdx.x * blockDim.x + threadIdx.x;
    if (i < n) p[i] = v;
}

// ---------------- degree accumulation ----------------
__global__ void deg_kernel(const int* __restrict__ row, const int* __restrict__ col,
                           float* __restrict__ deg, int E) {
    int e = blockIdx.x * blockDim.x + threadIdx.x;
    if (e >= E) return;
    int r = row[e], c = col[e];
    if (r != c) atomicAdd(&deg[r], 1.0f);
}

__global__ void rsqrt_kernel(float* __restrict__ deg, int n) {
    int i = blockIdx.x * blockDim.x + threadIdx.x;
    if (i < n) deg[i] = rsqrtf(deg[i]);
}

// ---------------- dense GEMM: h[N,64] = x[N,128] @ W[128,64] ----------------
// FP32 WMMA 16x16x4. One wave computes a 16x64 output strip (4 N-tiles),
// reusing the A fragment across the 4 tiles. All returns are wave-uniform so
// EXEC is all-ones at every WMMA.
__global__ void gemm_wmma_kernel(const float* __restrict__ x, const float* __restrict__ W,
                                 float* __restrict__ h, int nTiles) {
    const int wavesPerBlock = blockDim.x >> 5;
    const int wave = threadIdx.x >> 5;
    const int lane = threadIdx.x & 31;
    const int tile = blockIdx.x * wavesPerBlock + wave;   // wave-uniform
    if (tile >= nTiles) return;

    const int m0    = tile * 16;
    const int mr    = lane & 15;           // row within tile (A) / col within tile (B,C)
    const int halfK = (lane >> 4) << 1;    // 0 for lanes 0-15, 2 for lanes 16-31

    v8f acc[4] = {};                       // 4 N-tiles of 16x16 F32 accumulators

    const float* __restrict__ xrow = x + (long long)(m0 + mr) * 128;

    for (int k = 0; k < 128; k += 4) {
        // A fragment 16x4: lanes 0-15 -> K=k..k+1, lanes 16-31 -> K=k+2..k+3
        v2f a;
        a.x = xrow[k + halfK];
        a.y = xrow[k + halfK + 1];
#pragma unroll
        for (int t = 0; t < 4; ++t) {
            // B fragment 4x16: lower lanes hold first half of K range
            const int n = t * 16 + mr;
            v2f b;
            b.x = W[(k + halfK) * 64 + n];
            b.y = W[(k + halfK + 1) * 64 + n];
            acc[t] = __builtin_amdgcn_wmma_f32_16x16x4_f32(
                false, a, false, b, (short)0, acc[t], false, false);
        }
    }

    // C/D layout: VGPR v holds M = v + 8*(lane>=16), N = lane&15
    const int mhi = (lane >> 4) * 8;
#pragma unroll
    for (int t = 0; t < 4; ++t)
#pragma unroll
        for (int v = 0; v < 8; ++v)
            h[(long long)(m0 + mhi + v) * 64 + t * 16 + mr] = acc[t][v];
}

// Scalar tail for rows not covered by 16-row tiles (not launched for N=100000).
__global__ void gemm_tail_kernel(const float* __restrict__ x, const float* __restrict__ W,
                                 float* __restrict__ h, int mStart, int N) {
    int t = blockIdx.x * blockDim.x + threadIdx.x;
    int m = mStart + (t >> 6);
    int n = t & 63;
    if (m >= N) return;
    float s = 0.f;
    const float* xr = x + (long long)m * 128;
    for (int k = 0; k < 128; ++k) s += xr[k] * W[k * 64 + n];
    h[(long long)m * 64 + n] = s;
}

// ---------------- SpMM: dst[r] += dinv[r]*dinv[c] * src[c]  (r != c) ----------------
// 16 threads per edge, float4 per thread (64 floats per row).
__global__ void spmm_edges_kernel(const int* __restrict__ row, const int* __restrict__ col,
                                  const float* __restrict__ dinv,
                                  const float* __restrict__ src, float* __restrict__ dst,
                                  long long E) {
    long long t = (long long)blockIdx.x * blockDim.x + threadIdx.x;
    long long e = t >> 4;
    if (e >= E) return;
    const int f4 = (int)(t & 15);
    int r = row[e], c = col[e];
    if (r == c) return;                       // removed self loops
    float norm = dinv[r] * dinv[c];
    const float4 s = ((const float4*)(src + (long long)c * 64))[f4];
    float* d = dst + (long long)r * 64 + f4 * 4;
    atomicAdd(d + 0, norm * s.x);
    atomicAdd(d + 1, norm * s.y);
    atomicAdd(d + 2, norm * s.z);
    atomicAdd(d + 3, norm * s.w);
}

// Diagonal (added self loop): dst[i] += dinv[i]^2 * src[i]
__global__ void spmm_diag_kernel(const float* __restrict__ dinv,
                                 const float* __restrict__ src, float* __restrict__ dst,
                                 long long n4 /* = N*16 float4's */) {
    long long t = (long long)blockIdx.x * blockDim.x + threadIdx.x;
    if (t >= n4) return;
    int i = (int)(t >> 4);
    float s = dinv[i];
    s = s * s;
    float4 v = ((const float4*)src)[t];
    float4 o = ((float4*)dst)[t];
    o.x += s * v.x; o.y += s * v.y; o.z += s * v.z; o.w += s * v.w;
    ((float4*)dst)[t] = o;
}

// ---------------- launch ----------------
extern "C" void kernel_launch(void* const* d_in, const int* in_sizes, int n_in,
                              void* d_out, int out_size, void* d_ws, size_t ws_size,
                              hipStream_t stream) {
    const int*   edge = (const int*)d_in[0];     // [2,E] row-major: row then col
    const float* x    = (const float*)d_in[1];   // [N,128]
    const float* W    = (const float*)d_in[2];   // [128,64]
    float*       out  = (float*)d_out;           // [N,64]

    const int E = in_sizes[0] / 2;
    const int N = in_sizes[1] / 128;
    const int* row = edge;
    const int* col = edge + E;

    float* deg  = (float*)d_ws;                          // N floats (becomes dinv)
    float* hbuf = deg + ((long long)(N + 63) / 64) * 64; // N*64 floats, 256B aligned

    const int B = 256;

    // 1) deg = 1 (the added self loop)
    fill_kernel<<<(N + B - 1) / B, B, 0, stream>>>(deg, 1.0f, N);
    // 2) deg[r] += 1 per non-self edge
    deg_kernel<<<(E + B - 1) / B, B, 0, stream>>>(row, col, deg, E);
    // 3) deg -> deg^{-1/2}
    rsqrt_kernel<<<(N + B - 1) / B, B, 0, stream>>>(deg, N);

    // 4) out = x @ W via FP32 WMMA
    const int nTiles = N / 16;
    const int wavesPerBlock = B / 32;
    if (nTiles > 0)
        gemm_wmma_kernel<<<(nTiles + wavesPerBlock - 1) / wavesPerBlock, B, 0, stream>>>(
            x, W, out, nTiles);
    const int rem = N - nTiles * 16;
    if (rem > 0)
        gemm_tail_kernel<<<(rem * 64 + B - 1) / B, B, 0, stream>>>(x, W, out, nTiles * 16, N);

    const long long NF  = (long long)N * 64;  // elements of h
    const long long N4  = (long long)N * 16;  // float4's of h
    const long long ET  = (long long)E * 16;  // edge threads

    // 5) hbuf = 0 ; 6) hbuf = A_norm @ out
    fill_kernel<<<(int)((NF + B - 1) / B), B, 0, stream>>>(hbuf, 0.0f, NF);
    spmm_edges_kernel<<<(int)((ET + B - 1) / B), B, 0, stream>>>(row, col, deg, out, hbuf, E);
    spmm_diag_kernel<<<(int)((N4 + B - 1) / B), B, 0, stream>>>(deg, out, hbuf, N4);

    // 7) out = 0 ; 8) out = A_norm @ hbuf
    fill_kernel<<<(int)((NF + B - 1) / B), B, 0, stream>>>(out, 0.0f, NF);
    spmm_edges_kernel<<<(int)((ET + B - 1) / B), B, 0, stream>>>(row, col, deg, hbuf, out, E);
    spmm_diag_kernel<<<(int)((N4 + B - 1) / B), B, 0, stream>>>(deg, hbuf, out, N4);
}